// HyperConnection_52072183497069
// MI455X (gfx1250) — compile-verified
//
#include <hip/hip_runtime.h>

#define DIMX 2048
#define NS 4
#define NAL 5
#define NNN 2048
#define NB 4
#define NTILES (NB * NNN)   // 8192 (b,n) tiles
#define NWG 1024
#define EPS_F 1e-5f

typedef float    v4f   __attribute__((ext_vector_type(4)));
typedef unsigned u32x4 __attribute__((ext_vector_type(4)));
typedef unsigned u32x8 __attribute__((ext_vector_type(8)));

// Each wave issues ONE Tensor-Data-Mover descriptor moving its 4 KB half-row
// (row s = wave/2, half = wave&1) of the 32 KB (4 x 2048 f32) tile into LDS.
// 1D transfer: tile_dim0 = 1024 elements, data_size = 4 bytes.
// Issued unconditionally by every wave (TDM ignores EXEC; no branch needed),
// tracked per-wave with TENSORcnt.
__device__ __forceinline__ void issue_tile_tdm(const float* __restrict__ res,
                                               unsigned lds_base,
                                               int bufsel, int tile, int wave_u) {
  const int b = tile >> 11;            // tile / NNN
  const int n = tile & (NNN - 1);
  const int s = wave_u >> 1;
  const int half = wave_u & 1;

  const unsigned long gaddr = (unsigned long)res
      + (unsigned long)((unsigned)((b * NS + s) * NNN + n) * (DIMX * 4u)
                        + (unsigned)half * 4096u);
  const unsigned laddr = lds_base + (unsigned)bufsel * 32768u
                       + (unsigned)s * 8192u + (unsigned)half * 4096u;

  // D# group 0 (4 SGPRs): count=1 | lds_addr | global_addr[56:0] | type=2
  u32x4 g0;
  g0[0] = 1u;                                     // count = 1 valid descriptor
  g0[1] = laddr;                                  // LDS byte address
  g0[2] = (unsigned)gaddr;                        // global_addr[31:0]
  g0[3] = ((unsigned)(gaddr >> 32) & 0x01FFFFFFu) // global_addr[56:32]
        | 0x80000000u;                            // type = 2 ("image")

  // D# group 1 (8 SGPRs): mask=0, data_size=4B, dims for a 1024-elem 1D tile
  u32x8 g1;
  g1[0] = 0x00020000u;     // workgroup_mask=0, data_size=2 (4 bytes)
  g1[1] = 1024u << 16;     // tensor_dim0 = 1024 (bits 79:48, low half)
  g1[2] = 1u << 16;        // tensor_dim1 = 1    (bits 111:80, low half)
  g1[3] = 1024u << 16;     // tile_dim0  = 1024  (bits 127:112)
  g1[4] = 1u;              // tile_dim1  = 1, tile_dim2 = 0
  g1[5] = 1024u;           // tensor_dim0_stride (bits 207:160, low 32)
  g1[6] = 0u;
  g1[7] = 0u;

  asm volatile("tensor_load_to_lds %0, %1" :: "s"(g0), "s"(g1) : "memory");
}

extern "C" __global__ void __launch_bounds__(256)
hc_kernel(const float* __restrict__ res,      // (B*S, N, DIM) f32
          const float* __restrict__ nw,       // (DIM,)
          const float* __restrict__ sbeta,    // (S,)
          const float* __restrict__ salpha,   // (S, S+V) = (4,5)
          const float* __restrict__ fa,       // (DIM, 5)
          const float* __restrict__ das_p,    // scalar
          const float* __restrict__ fb,       // (DIM,)
          const float* __restrict__ dbs_p,    // scalar
          float* __restrict__ out)            // (B*S, N, DIM)
{
  __shared__ float smem[2 * NS * DIMX];       // exactly 64 KB: two 32 KB tile buffers

  const int tid  = threadIdx.x;
  const int lane = tid & 31;
  const int wave = tid >> 5;
  // wave id as a compiler-visible scalar so descriptor math stays in SALU
  const int wave_u = __builtin_amdgcn_readfirstlane(tid >> 5);
  const unsigned lds_base = (unsigned)(size_t)(&smem[0]);

  const int tile0 = blockIdx.x;
  // kick off the first tile's TDM copy before doing any setup work
  issue_tile_tdm(res, lds_base, 0, tile0, wave_u);

  // per-thread products w[d]*fa[d][j] and w[d]*fb[d] for this thread's 8 d-values
  float wfa[NAL][8];
  float wfb[8];
  {
    const int d0 = tid * 8;
#pragma unroll
    for (int k = 0; k < 8; ++k) {
      const float wd = nw[d0 + k];
      wfb[k] = wd * fb[d0 + k];
#pragma unroll
      for (int j = 0; j < NAL; ++j)
        wfa[j][k] = wd * fa[(d0 + k) * NAL + j];
    }
  }
  const float das = das_p[0];
  const float dbs = dbs_p[0];

  // cw[j] = sum_d w*fa_j (j<5), cw[5] = sum_d w*fb  -- mean-correction constants.
  // Cross-wave reduce through buffer 1 (no TDM has touched it yet).
  float cw[6];
  {
    float c6[6];
#pragma unroll
    for (int j = 0; j < NAL; ++j) {
      float a = 0.f;
#pragma unroll
      for (int k = 0; k < 8; ++k) a += wfa[j][k];
      c6[j] = a;
    }
    {
      float a = 0.f;
#pragma unroll
      for (int k = 0; k < 8; ++k) a += wfb[k];
      c6[5] = a;
    }
#pragma unroll
    for (int j = 0; j < 6; ++j)
#pragma unroll
      for (int m = 16; m >= 1; m >>= 1)
        c6[j] += __shfl_xor(c6[j], m, 32);
    float* scr = smem + NS * DIMX;            // buffer 1 as one-time scratch
    if (lane == 0) {
#pragma unroll
      for (int j = 0; j < 6; ++j) scr[wave * 8 + j] = c6[j];
    }
    __syncthreads();
#pragma unroll
    for (int j = 0; j < 6; ++j) {
      float a = 0.f;
#pragma unroll
      for (int w8 = 0; w8 < 8; ++w8) a += scr[w8 * 8 + j];
      cw[j] = a;
    }
    // loop-top barrier below orders these reads before tile1's TDM writes
  }

  int bufsel = 0;
  for (int tile = tile0; tile < NTILES; tile += NWG) {
    __builtin_amdgcn_s_wait_tensorcnt(0);   // this wave's slice landed
    __syncthreads();                        // whole tile landed + prev compute done

    const int nxt = tile + NWG;
    if (nxt < NTILES)
      issue_tile_tdm(res, lds_base, bufsel ^ 1, nxt, wave_u);

    float* buf = smem + bufsel * (NS * DIMX);

    // pull this thread's 4x8 tile slice into registers
    float xv[NS][8];
#pragma unroll
    for (int s = 0; s < NS; ++s) {
      const v4f a = *(const v4f*)(buf + s * DIMX + tid * 8);
      const v4f b = *(const v4f*)(buf + s * DIMX + tid * 8 + 4);
      xv[s][0] = a.x; xv[s][1] = a.y; xv[s][2] = a.z; xv[s][3] = a.w;
      xv[s][4] = b.x; xv[s][5] = b.y; xv[s][6] = b.z; xv[s][7] = b.w;
    }

    // 32 partial sums per thread: [s*8+q], q: 0=Σx, 1=Σx², 2..6=Σx·wfa_j, 7=Σx·wfb
    float acc[32];
#pragma unroll
    for (int i = 0; i < 32; ++i) acc[i] = 0.f;
#pragma unroll
    for (int s = 0; s < NS; ++s) {
#pragma unroll
      for (int k = 0; k < 8; ++k) {
        const float x = xv[s][k];
        acc[s * 8 + 0] += x;
        acc[s * 8 + 1] = fmaf(x, x, acc[s * 8 + 1]);
#pragma unroll
        for (int j = 0; j < NAL; ++j)
          acc[s * 8 + 2 + j] = fmaf(x, wfa[j][k], acc[s * 8 + 2 + j]);
        acc[s * 8 + 7] = fmaf(x, wfb[k], acc[s * 8 + 7]);
      }
    }

    // multi-value wave32 butterfly: 31 shuffles reduce all 32 sums;
    // lane L ends holding the wave-total of scalar index L in acc[0].
#pragma unroll
    for (int m = 16; m >= 1; m >>= 1) {
#pragma unroll
      for (int i = 0; i < m; ++i) {
        const float send = (lane & m) ? acc[i] : acc[i + m];
        const float recv = __shfl_xor(send, m, 32);
        const float keep = (lane & m) ? acc[i + m] : acc[i];
        acc[i] = keep + recv;
      }
    }

    __syncthreads();              // all threads finished reading buf -> overlay scratch
    float* red  = buf;            // [8][32] per-wave partials
    float* scal = buf + 256;      // [32] workgroup totals
    float* alf  = buf + 288;      // [4][5] alpha
    float* bet  = buf + 308;      // [4] beta

    red[wave * 32 + lane] = acc[0];
    __syncthreads();
    if (tid < 32) {
      float a = 0.f;
#pragma unroll
      for (int w8 = 0; w8 < 8; ++w8) a += red[w8 * 32 + tid];
      scal[tid] = a;
    }
    __syncthreads();
    if (tid < NS) {               // 4 threads compute the 24 tanh gates
      const int s = tid;
      const float mu  = scal[s * 8 + 0] * (1.0f / DIMX);
      const float var = scal[s * 8 + 1] * (1.0f / DIMX) - mu * mu;
      const float inv = rsqrtf(var + EPS_F);
#pragma unroll
      for (int j = 0; j < NAL; ++j) {
        const float dot = inv * (scal[s * 8 + 2 + j] - mu * cw[j]);
        alf[s * NAL + j] = tanhf(dot) * das + salpha[s * NAL + j];
      }
      const float dotb = inv * (scal[s * 8 + 7] - mu * cw[5]);
      bet[s] = tanhf(dotb) * dbs + sbeta[s];
    }
    __syncthreads();

    // out[s][d] = Σ_s' C[s][s'] x[s'][d],  C[s][s'] = beta[s]*alf[s'][0] + alf[s'][s+1]
    float C[NS][NS];
#pragma unroll
    for (int s = 0; s < NS; ++s) {
      const float bs = bet[s];
#pragma unroll
      for (int sp = 0; sp < NS; ++sp)
        C[s][sp] = fmaf(bs, alf[sp * NAL + 0], alf[sp * NAL + (s + 1)]);
    }

    const int b = tile >> 11;
    const int n = tile & (NNN - 1);
#pragma unroll
    for (int s = 0; s < NS; ++s) {
      float o[8];
#pragma unroll
      for (int k = 0; k < 8; ++k) {
        float v = C[s][0] * xv[0][k];
        v = fmaf(C[s][1], xv[1][k], v);
        v = fmaf(C[s][2], xv[2][k], v);
        v = fmaf(C[s][3], xv[3][k], v);
        o[k] = v;
      }
      const size_t off = ((size_t)((b * NS + s) * NNN + n)) * DIMX + (size_t)tid * 8;
      v4f o0 = {o[0], o[1], o[2], o[3]};
      v4f o1 = {o[4], o[5], o[6], o[7]};
      __builtin_nontemporal_store(o0, (v4f*)(out + off));
      __builtin_nontemporal_store(o1, (v4f*)(out + off + 4));
    }

    bufsel ^= 1;
  }
}

extern "C" void kernel_launch(void* const* d_in, const int* in_sizes, int n_in,
                              void* d_out, int out_size, void* d_ws, size_t ws_size,
                              hipStream_t stream) {
  (void)in_sizes; (void)n_in; (void)out_size; (void)d_ws; (void)ws_size;
  hc_kernel<<<dim3(NWG), dim3(256), 0, stream>>>(
      (const float*)d_in[0],   // residuals
      (const float*)d_in[1],   // norm_weight
      (const float*)d_in[2],   // static_beta
      (const float*)d_in[3],   // static_alpha
      (const float*)d_in[4],   // dynamic_alpha_fn
      (const float*)d_in[5],   // dynamic_alpha_scale
      (const float*)d_in[6],   // dynamic_beta_fn
      (const float*)d_in[7],   // dynamic_beta_scale
      (float*)d_out);
}